// ALDC_ISTA_40252433498286
// MI455X (gfx1250) — compile-verified
//
#include <hip/hip_runtime.h>
#include <hip/hip_bf16.h>

// ---------------------------------------------------------------------------
// ALDC-ISTA on MI455X (gfx1250).
//   yW2 = y @ W2^T            (4096x512x2048)   ~8.6 GFLOP
//   5x:  g = x - mu*(x@W1^T - yW2) + la*ddx_exp(x)   (4096x2048x2048 each)
//   6x:  per-row top-k soft-threshold (radix select, exact tie semantics)
// All f32; compute-bound on the matrix pipe -> V_WMMA_F32_16X16X4_F32.
// Working set (~92MB) is L2-resident (192MB), so LDS-tiled WMMA GEMM with a
// fused epilogue is the whole game.
// ---------------------------------------------------------------------------

typedef float v2f __attribute__((ext_vector_type(2)));
typedef float v8f __attribute__((ext_vector_type(8)));

#define B_ROWS 4096
#define NCOLS  2048
#define TILE_M 128
#define TILE_N 128
#define TILE_K 16
#define LDSPAD 20   // 80B rows: float4 LDS stores stay 16B aligned; stride-20
                    // frag reads hit all 64 banks exactly once (conflict-free)

// D = A(16x4) x B(4x16) + C fragment indexing (wave32):
//   A vgpr0: lanes0-15 K=0, lanes16-31 K=2 ; vgpr1: K=1 / K=3  (M = lane%16)
//   B mirrors A with N = lane%16; C vgpr e: M = e + 8*(lane>=16), N = lane%16
__global__ __launch_bounds__(256, 2)
void gemm_wmma_f32_fused(const float* __restrict__ A,     // [rows x K] row-major
                         const float* __restrict__ W,     // [NCOLS x K] row-major; B[k][n] = W[n][k]
                         float* __restrict__ out,         // [rows x NCOLS]
                         const float* __restrict__ xin,   // mode 1: previous x
                         const float* __restrict__ yW2,   // mode 1: yW2
                         const float* __restrict__ mu_p,
                         const float* __restrict__ lambd_p,
                         const float* __restrict__ theta_p,
                         int K, int mode, int t)
{
  __shared__ float Ads[TILE_M][LDSPAD];
  __shared__ float Bds[TILE_N][LDSPAD];

  const int tid  = threadIdx.x;
  const int lane = tid & 31;
  const int lh   = lane >> 4;       // lane half (0/1)
  const int l16  = lane & 15;
  const int wave = tid >> 5;        // 8 waves
  const int wm   = wave >> 1;       // 0..3 -> 32-row strip
  const int wn   = wave & 1;        // 0..1 -> 64-col strip
  const int m0   = blockIdx.y * TILE_M;
  const int n0   = blockIdx.x * TILE_N;

  v8f acc[2][4] = {};

  // Tile-fill assignment: 256 threads x 2 float4 per operand tile.
  const int ldr = tid >> 2;         // 0..63
  const int ldc = (tid & 3) * 4;    // 0,4,8,12

  for (int ks = 0; ks < K; ks += TILE_K) {
    // Issue global loads early (overlap previous slab's WMMAs via barrier).
    float4 a0 = *(const float4*)(A + (size_t)(m0 + ldr)      * K + ks + ldc);
    float4 a1 = *(const float4*)(A + (size_t)(m0 + ldr + 64) * K + ks + ldc);
    float4 b0 = *(const float4*)(W + (size_t)(n0 + ldr)      * K + ks + ldc);
    float4 b1 = *(const float4*)(W + (size_t)(n0 + ldr + 64) * K + ks + ldc);
    __syncthreads();                              // LDS free to overwrite
    *(float4*)&Ads[ldr][ldc]      = a0;
    *(float4*)&Ads[ldr + 64][ldc] = a1;
    *(float4*)&Bds[ldr][ldc]      = b0;           // Bds[n][k]
    *(float4*)&Bds[ldr + 64][ldc] = b1;
    __syncthreads();

#pragma unroll
    for (int kk = 0; kk < TILE_K; kk += 4) {
      v2f af[2], bf[4];
#pragma unroll
      for (int i = 0; i < 2; ++i)
        af[i] = *(const v2f*)&Ads[wm * 32 + i * 16 + l16][kk + 2 * lh];
#pragma unroll
      for (int j = 0; j < 4; ++j)
        bf[j] = *(const v2f*)&Bds[wn * 64 + j * 16 + l16][kk + 2 * lh];
#pragma unroll
      for (int i = 0; i < 2; ++i)
#pragma unroll
        for (int j = 0; j < 4; ++j)
          acc[i][j] = __builtin_amdgcn_wmma_f32_16x16x4_f32(
              false, af[i], false, bf[j], (short)0, acc[i][j], false, false);
    }
  }

  if (mode == 0) {
#pragma unroll
    for (int i = 0; i < 2; ++i)
#pragma unroll
      for (int j = 0; j < 4; ++j) {
        const int col = n0 + wn * 64 + j * 16 + l16;
#pragma unroll
        for (int e = 0; e < 8; ++e) {
          const int row = m0 + wm * 32 + i * 16 + lh * 8 + e;
          out[(size_t)row * NCOLS + col] = acc[i][j][e];
        }
      }
  } else {
    // Fused gradient-step epilogue: g = x - mu*(acc - yW2) + la*ddx_exp(x,th)
    const float mu = mu_p[t], la = lambd_p[t], th = theta_p[t];
#pragma unroll
    for (int i = 0; i < 2; ++i)
#pragma unroll
      for (int j = 0; j < 4; ++j) {
        const int col = n0 + wn * 64 + j * 16 + l16;
#pragma unroll
        for (int e = 0; e < 8; ++e) {
          const int row = m0 + wm * 32 + i * 16 + lh * 8 + e;
          const size_t idx = (size_t)row * NCOLS + col;
          const float xv = xin[idx];
          const float yv = yW2[idx];
          const float dd = __builtin_copysignf(
              th * (1.0f - __expf(-th * __builtin_fabsf(xv))), xv);
          out[idx] = xv - mu * (acc[i][j][e] - yv) + la * dd;
        }
      }
  }
}

// ---------------------------------------------------------------------------
// Per-row top-k soft threshold. One 256-thread WG per row of 2048.
// Exact k-th-largest-|v| via 4-pass MSD radix select on float bits (|v| >= 0
// so bit order == value order). mask = |v| >= thr matches reference tie
// semantics (abs_x >= sorted_abs[idx-1]) exactly.
// ---------------------------------------------------------------------------
__global__ __launch_bounds__(256)
void shrink_topk(const float* __restrict__ g, float* __restrict__ xout,
                 const float* __restrict__ lambd_p,
                 const float* __restrict__ theta_p,
                 const float* __restrict__ mu_p,
                 int pidx, int scale_idx, int k)
{
  __shared__ unsigned hist[256];
  __shared__ unsigned sh_pref;
  __shared__ unsigned sh_rem;

  const int row = blockIdx.x;
  const int tid = threadIdx.x;
  const float scale = (scale_idx >= 0) ? mu_p[scale_idx] : 1.0f;
  const float beta  = theta_p[pidx] * lambd_p[pidx];

  const float* gr = g + (size_t)row * NCOLS;
  float v[8];
  unsigned ab[8];
#pragma unroll
  for (int e = 0; e < 8; ++e) {
    const float val = gr[tid + e * 256] * scale;
    v[e]  = val;
    ab[e] = __float_as_uint(__builtin_fabsf(val));
  }

  if (tid == 0) { sh_pref = 0u; sh_rem = (unsigned)k; }

  for (int p = 3; p >= 0; --p) {
    hist[tid] = 0u;
    __syncthreads();                 // hist zeroed; sh_pref/sh_rem visible
    const unsigned shift = (unsigned)(p * 8);
    const unsigned umask = (p == 3) ? 0u : (0xFFFFFFFFu << (shift + 8));
    const unsigned pref  = sh_pref;
#pragma unroll
    for (int e = 0; e < 8; ++e)
      if ((ab[e] & umask) == pref)
        atomicAdd(&hist[(ab[e] >> shift) & 255u], 1u);
    __syncthreads();
    if (tid == 0) {
      unsigned rem = sh_rem, cum = 0;
      int b;
      for (b = 255; b >= 0; --b) { cum += hist[b]; if (cum >= rem) break; }
      if (b < 0) b = 0;
      sh_rem  = rem - (cum - hist[b]);
      sh_pref = pref | ((unsigned)b << shift);
    }
    __syncthreads();
  }
  const unsigned thr_bits = sh_pref;

  float* xr = xout + (size_t)row * NCOLS;
#pragma unroll
  for (int e = 0; e < 8; ++e) {
    const float val  = v[e];
    const float soft = __builtin_copysignf(
        fmaxf(__builtin_fabsf(val) - beta, 0.0f), val);
    xr[tid + e * 256] = (ab[e] >= thr_bits) ? val : soft;
  }
}

// ---------------------------------------------------------------------------
// Host orchestration. Inputs: y(4096x512), W1(2048x2048), W2(2048x512),
// lambd_p(6), mu_p(6), theta_p(6). Output: x(4096x2048) f32 in d_out.
// Scratch: yW2 (32MB) + grad (32MB) in d_ws.
// ---------------------------------------------------------------------------
extern "C" void kernel_launch(void* const* d_in, const int* in_sizes, int n_in,
                              void* d_out, int out_size, void* d_ws, size_t ws_size,
                              hipStream_t stream) {
  const float* y      = (const float*)d_in[0];
  const float* W1     = (const float*)d_in[1];
  const float* W2     = (const float*)d_in[2];
  const float* lambd  = (const float*)d_in[3];
  const float* mu     = (const float*)d_in[4];
  const float* theta  = (const float*)d_in[5];
  float* x    = (float*)d_out;
  float* yW2  = (float*)d_ws;
  float* grad = (float*)d_ws + (size_t)B_ROWS * NCOLS;

  const dim3 ggrid(NCOLS / TILE_N, B_ROWS / TILE_M);   // (16, 32)

  // yW2 = y @ W2^T   (K = 512)
  gemm_wmma_f32_fused<<<ggrid, 256, 0, stream>>>(
      y, W2, yW2, yW2, yW2, mu, lambd, theta, 512, 0, 0);

  // x = shrink(mu[0] * yW2, theta[0]*lambd[0], t=1)
  shrink_topk<<<B_ROWS, 256, 0, stream>>>(
      yW2, x, lambd, theta, mu, /*pidx=*/0, /*scale_idx=*/0,
      (int)(0.012 * (double)NCOLS));                   // 24

  for (int t = 1; t <= 5; ++t) {
    // grad = x - mu[t]*(x@W1^T - yW2) + lambd[t]*ddx_exp(x, theta[t])
    gemm_wmma_f32_fused<<<ggrid, 256, 0, stream>>>(
        x, W1, grad, x, yW2, mu, lambd, theta, 2048, 1, t);
    double p = 0.012 * (double)t;
    if (p > 0.12) p = 0.12;
    const int k = (int)(p * (double)NCOLS);            // 24,49,73,98,122
    shrink_topk<<<B_ROWS, 256, 0, stream>>>(
        grad, x, lambd, theta, mu, /*pidx=*/t, /*scale_idx=*/-1, k);
  }
}